// O3TensorProduct_7868380086394
// MI455X (gfx1250) — compile-verified
//
#include <hip/hip_runtime.h>

typedef float v2f __attribute__((ext_vector_type(2)));
typedef float v4f __attribute__((ext_vector_type(4)));
typedef float v8f __attribute__((ext_vector_type(8)));

#define MULC  128
#define ROWS  32          // samples per block (two 16-row M tiles per wave)
#define LDWZ  260         // 256 + 4 pad words (1040 B row: 16B-aligned, banks 4r+c)
#define LDW1  388         // 384 + 4 pad words (1552 B row: 16B-aligned, banks 4r+c)

// ---- CDNA5 async global->LDS copy (ASYNCcnt-tracked, no VGPR round-trip) ----
__device__ __forceinline__ void async_copy8(unsigned lds_off, const void* gptr) {
    asm volatile("global_load_async_to_lds_b64 %0, %1, off"
                 :: "v"(lds_off), "v"(gptr)
                 : "memory");
}
__device__ __forceinline__ void wait_async0() {
    asm volatile("s_wait_asynccnt 0x0" ::: "memory");
}
// LDS aperture keeps the wave-relative offset in addr[31:0]
__device__ __forceinline__ unsigned lds_off32(const void* p) {
    return (unsigned)(uintptr_t)p;
}

// ---------------------------------------------------------------------------
// One-time W repack into WMMA B-fragment order:
//   entry (k2 = 2k+hi, col) = { W000[kk][col], W000[kk+1][col],
//                               W110[kk][col], W110[kk+1][col],
//                               W011[kk][col], W011[kk+1][col],
//                               W101[kk][col], W101[kk+1][col] }, kk = 2*k2.
//   A lane then gets all four B-frags with two b128 loads.
// ---------------------------------------------------------------------------
__global__ __launch_bounds__(256)
void repack_w_kernel(const float* __restrict__ W000, const float* __restrict__ W110,
                     const float* __restrict__ W011, const float* __restrict__ W101,
                     float* __restrict__ wp)
{
    int idx = blockIdx.x * 256 + threadIdx.x;    // 64 * 128 entries
    if (idx >= 64 * 128) return;
    int k2  = idx >> 7;
    int col = idx & 127;
    int kk  = k2 * 2;
    float* dst = wp + (size_t)idx * 8;
    dst[0] = W000[kk * MULC + col]; dst[1] = W000[(kk + 1) * MULC + col];
    dst[2] = W110[kk * MULC + col]; dst[3] = W110[(kk + 1) * MULC + col];
    dst[4] = W011[kk * MULC + col]; dst[5] = W011[(kk + 1) * MULC + col];
    dst[6] = W101[kk * MULC + col]; dst[7] = W101[(kk + 1) * MULC + col];
}

template <bool PACKED>
__global__ __launch_bounds__(256)
void o3tp_kernel(const float* __restrict__ x0,
                 const float* __restrict__ x1,
                 const float* __restrict__ y0,
                 const float* __restrict__ y1,
                 const float* __restrict__ W000,
                 const float* __restrict__ W110,
                 const float* __restrict__ W011,
                 const float* __restrict__ W101,
                 const float* __restrict__ bias,
                 const float* __restrict__ wpack,
                 float* __restrict__ out)
{
    // Pair-major: word (r, p*4 + {0,1}) = x0[r][2p], x0[r][2p+1]
    //             word (r, p*4 + {2,3}) = z [r][2p], z [r][2p+1]
    __shared__ alignas(16) float sxz[ROWS][LDWZ];
    // Fragment-major: word (r, p*6 + i*2 + j) = x1[r][2p+j][i]
    __shared__ alignas(16) float sx1p[ROWS][LDW1];
    __shared__ float sy0[ROWS];
    __shared__ float sy1[ROWS][3];

    const int tid  = threadIdx.x;
    const int row0 = blockIdx.x * ROWS;

    // ---- phase A: async-stage x0 pairs; load y -> LDS ----
    {
        const float* src = x0 + (size_t)row0 * MULC;
#pragma unroll
        for (int q = 0; q < 8; ++q) {
            int c = tid + 256 * q;                 // pair id, 64 pairs/row
            int r = c >> 6, p = c & 63;
            async_copy8(lds_off32(&sxz[r][p * 4]), src + (size_t)c * 2);
        }
    }
    if (tid < ROWS) {
        int n = row0 + tid;
        sy0[tid]    = y0[n];
        sy1[tid][0] = y1[n * 3 + 0];
        sy1[tid][1] = y1[n * 3 + 1];
        sy1[tid][2] = y1[n * 3 + 2];
    }
    __syncthreads();   // sy1 visible for phase B

    // ---- phase B: x1 from global (coalesced b96), z + swizzled scatter ----
    {
        const float* src = x1 + (size_t)row0 * MULC * 3;
#pragma unroll
        for (int q = 0; q < 16; ++q) {
            int j = tid + 256 * q;                 // (r, u)
            int r = j >> 7, u = j & 127;
            const float* pv = src + (size_t)r * 384 + 3 * u;
            float v0i = pv[0], v1i = pv[1], v2i = pv[2];
            float zz = v0i * sy1[r][0] + v1i * sy1[r][1] + v2i * sy1[r][2];
            int p = u >> 1, jj = u & 1;
            float* d = &sx1p[r][p * 6 + jj];
            d[0] = v0i; d[2] = v1i; d[4] = v2i;
            sxz[r][p * 4 + 2 + jj] = zz;
        }
    }
    wait_async0();
    __syncthreads();

    // ---- per-wave: one 16-col tile, two 16-row M tiles, 12 accumulators ----
    const int wave   = tid >> 5;      // 0..7 -> column tile
    const int lane   = tid & 31;
    const int hi     = lane >> 4;     // K half for A/B frags, M half for C/D
    const int lanelo = lane & 15;
    const int col    = wave * 16 + lanelo;

    v8f accS0[2] = {}, accS1[2] = {}, accC[2] = {},
        accD0[2] = {}, accD1[2] = {}, accD2[2] = {};

    const float* wpk = PACKED ? (wpack + ((size_t)hi * MULC + col) * 8) : nullptr;

#pragma unroll 2
    for (int k = 0; k < 32; ++k) {
        const int p = 2 * k + hi;        // K-pair index; kk = 2p

        // B-frags (4x16 f32 layout): N = lanelo, rows K = 2p, 2p+1
        v2f b000, b110, b011, b101;
        if (PACKED) {
            const float* wk = wpk + (size_t)k * 2048;  // k stride: 2*128*8 floats
            v4f q0 = *(const v4f*)(wk);
            v4f q1 = *(const v4f*)(wk + 4);
            b000.x = q0.x; b000.y = q0.y;
            b110.x = q0.z; b110.y = q0.w;
            b011.x = q1.x; b011.y = q1.y;
            b101.x = q1.z; b101.y = q1.w;
        } else {
            const int idx = 2 * p * MULC + col;
            b000.x = W000[idx]; b000.y = W000[idx + MULC];
            b110.x = W110[idx]; b110.y = W110[idx + MULC];
            b011.x = W011[idx]; b011.y = W011[idx + MULC];
            b101.x = W101[idx]; b101.y = W101[idx + MULC];
        }

#pragma unroll
        for (int t = 0; t < 2; ++t) {
            const int rw = t * 16 + lanelo;
            // one b128: {x0[2p], x0[2p+1], z[2p], z[2p+1]} -> two even pairs
            v4f qa = *(const v4f*)&sxz[rw][p * 4];
            v2f ax0 = { qa.x, qa.y };
            v2f az  = { qa.z, qa.w };
            // six consecutive words -> three b64 fragments, no regrouping
            const float* xf = &sx1p[rw][p * 6];
            v2f a0 = { xf[0], xf[1] };
            v2f a1 = { xf[2], xf[3] };
            v2f a2 = { xf[4], xf[5] };

            accS0[t] = __builtin_amdgcn_wmma_f32_16x16x4_f32(false, ax0, false, b000, (short)0, accS0[t], false, false);
            accS1[t] = __builtin_amdgcn_wmma_f32_16x16x4_f32(false, az,  false, b110, (short)0, accS1[t], false, false);
            accC [t] = __builtin_amdgcn_wmma_f32_16x16x4_f32(false, ax0, false, b011, (short)0, accC [t], false, false);
            accD0[t] = __builtin_amdgcn_wmma_f32_16x16x4_f32(false, a0,  false, b101, (short)0, accD0[t], false, false);
            accD1[t] = __builtin_amdgcn_wmma_f32_16x16x4_f32(false, a1,  false, b101, (short)0, accD1[t], false, false);
            accD2[t] = __builtin_amdgcn_wmma_f32_16x16x4_f32(false, a2,  false, b101, (short)0, accD2[t], false, false);
        }
    }

    // ---- epilogue: C/D layout -> VGPR r holds (M = r + 8*hi, N = lanelo) ----
    const float inv_sqrt3 = 0.57735026918962576f;
    const float bc = bias[col];

#pragma unroll
    for (int t = 0; t < 2; ++t) {
#pragma unroll
        for (int r = 0; r < 8; ++r) {
            const int m = t * 16 + r + 8 * hi;
            const size_t n = (size_t)(row0 + m);
            const float Y0 = sy0[m];

            // scalar channel: (x0@W000)*y0 + (z@W110)/sqrt3 + b
            float s = accS0[t][r] * Y0 + accS1[t][r] * inv_sqrt3 + bc;
            __builtin_nontemporal_store(s, &out[n * 512 + col]);

            // vector channel: ((x0@W011)*y1_i + (x1_i@W101)*y0) / sqrt3
            const size_t vb = n * 512 + 128 + (size_t)col * 3;
            float v0 = (accC[t][r] * sy1[m][0] + accD0[t][r] * Y0) * inv_sqrt3;
            float v1 = (accC[t][r] * sy1[m][1] + accD1[t][r] * Y0) * inv_sqrt3;
            float v2 = (accC[t][r] * sy1[m][2] + accD2[t][r] * Y0) * inv_sqrt3;
            __builtin_nontemporal_store(v0, &out[vb + 0]);
            __builtin_nontemporal_store(v1, &out[vb + 1]);
            __builtin_nontemporal_store(v2, &out[vb + 2]);
        }
    }
}

extern "C" void kernel_launch(void* const* d_in, const int* in_sizes, int n_in,
                              void* d_out, int out_size, void* d_ws, size_t ws_size,
                              hipStream_t stream) {
    const float* x0   = (const float*)d_in[0];
    const float* x1   = (const float*)d_in[1];
    const float* y0   = (const float*)d_in[2];
    const float* y1   = (const float*)d_in[3];
    const float* W000 = (const float*)d_in[4];
    const float* W110 = (const float*)d_in[5];
    const float* W011 = (const float*)d_in[6];
    const float* W101 = (const float*)d_in[7];
    const float* b    = (const float*)d_in[8];
    float* out = (float*)d_out;

    const int N      = in_sizes[0] / MULC;   // 100000
    const int blocks = N / ROWS;             // 3125 (N % 32 == 0)

    const size_t packed_bytes = (size_t)64 * 128 * 8 * sizeof(float);  // 256 KB
    if (ws_size >= packed_bytes) {
        float* wp = (float*)d_ws;
        repack_w_kernel<<<dim3(32), dim3(256), 0, stream>>>(W000, W110, W011, W101, wp);
        o3tp_kernel<true><<<dim3(blocks), dim3(256), 0, stream>>>(
            x0, x1, y0, y1, W000, W110, W011, W101, b, wp, out);
    } else {
        o3tp_kernel<false><<<dim3(blocks), dim3(256), 0, stream>>>(
            x0, x1, y0, y1, W000, W110, W011, W101, b, nullptr, out);
    }
}